// DGCNN_partseg_8761733284519
// MI455X (gfx1250) — compile-verified
//
#include <hip/hip_runtime.h>
#include <hip/hip_bf16.h>
#include <math.h>

// ---------------------------------------------------------------------------
// DGCNN part-seg forward for gfx1250 (MI455X), compile-only loop.
// All GEMMs (KNN Gram, conv1..5, fc, center convs) use V_WMMA_F32_16X16X4_F32.
// Contraction dims are pre-padded to multiples of 4 so the WMMA K-loops are
// straight-line (no exec-mask toggling). Each wave computes a 16x64 output
// tile (4 WMMAs per K-step, A-fragment reuse). BatchNorm is two-pass (WMMA
// stats pass + WMMA epilogue pass); max-over-k folds through BN+leaky
// (monotone, a>0) via atomic float-max on raw conv output.
// ---------------------------------------------------------------------------

typedef __attribute__((ext_vector_type(2))) float v2f;
typedef __attribute__((ext_vector_type(8))) float v8f;

static inline int ceilDiv(int a, int b) { return (a + b - 1) / b; }

__device__ __forceinline__ float leaky02(float y) { return y >= 0.f ? y : 0.2f * y; }

__device__ __forceinline__ void atomicMaxF(float* addr, float val) {
  int* ia = (int*)addr;
  int cur = __float_as_int(*addr);
  while (__int_as_float(cur) < val) {
    int prev = atomicCAS(ia, cur, __float_as_int(val));
    if (prev == cur) break;
    cur = prev;
  }
}

// ---------------------------------------------------------------------------
__global__ void fill_f32(float* __restrict__ p, float v, long n) {
  long t = (long)blockIdx.x * blockDim.x + threadIdx.x;
  if (t < n) p[t] = v;
}

// pad conv1 weights (64,6) -> (64,8) with zeros. 512 threads.
__global__ void pad_conv1w(const float* __restrict__ W, float* __restrict__ Wp) {
  int t = threadIdx.x;
  int o = t >> 3, c = t & 7;
  Wp[o * 8 + c] = (c < 6) ? W[o * 6 + c] : 0.f;
}

// pad x (b,3,N) -> (b,4,N) with zero channel 3 (for guard-free Gram).
__global__ void pad_x4(const float* __restrict__ x, float* __restrict__ xp, int Npts) {
  int t = blockIdx.x * blockDim.x + threadIdx.x;
  int b = blockIdx.y;
  if (t >= 4 * Npts) return;
  int c = t / Npts, n = t % Npts;
  xp[((size_t)b * 4 + c) * Npts + n] = (c < 3) ? x[((size_t)b * 3 + c) * Npts + n] : 0.f;
}

// ---------------------------------------------------------------------------
// Gram matrix G[n,m] = sum_c X[c,n]*X[c,m] for one batch. C % 4 == 0.
// Wave computes 16 rows x 64 cols. Grid: (Npts/64, Npts/16), 32 threads.
__global__ __launch_bounds__(32) void gram_wmma(const float* __restrict__ X,
                                                float* __restrict__ G,
                                                int C, int Npts) {
  const int lane = threadIdx.x & 31;
  const int half = lane >> 4, l16 = lane & 15;
  const int tn = blockIdx.x * 64;  // columns (m)
  const int tm = blockIdx.y * 16;  // rows (n)
  const v8f vz = {};
  v8f accs[4] = {vz, vz, vz, vz};
  const float* pa = X + (size_t)(2 * half) * Npts + tm + l16;
  const float* pb = X + (size_t)(2 * half) * Npts + tn + l16;
  const int ksteps = C >> 2;
  for (int t = 0; t < ksteps; ++t) {
    v2f a;
    a.x = pa[0];
    a.y = pa[Npts];
    v2f b0, b1, b2, b3;
    b0.x = pb[0];  b0.y = pb[Npts];
    b1.x = pb[16]; b1.y = pb[Npts + 16];
    b2.x = pb[32]; b2.y = pb[Npts + 32];
    b3.x = pb[48]; b3.y = pb[Npts + 48];
    accs[0] = __builtin_amdgcn_wmma_f32_16x16x4_f32(false, a, false, b0, (short)0, accs[0], false, false);
    accs[1] = __builtin_amdgcn_wmma_f32_16x16x4_f32(false, a, false, b1, (short)0, accs[1], false, false);
    accs[2] = __builtin_amdgcn_wmma_f32_16x16x4_f32(false, a, false, b2, (short)0, accs[2], false, false);
    accs[3] = __builtin_amdgcn_wmma_f32_16x16x4_f32(false, a, false, b3, (short)0, accs[3], false, false);
    pa += (size_t)4 * Npts;
    pb += (size_t)4 * Npts;
  }
#pragma unroll
  for (int nt = 0; nt < 4; ++nt)
#pragma unroll
    for (int v = 0; v < 8; ++v) {
      int rm = tm + v + 8 * half;
      G[(size_t)rm * Npts + tn + nt * 16 + l16] = accs[nt][v];
    }
}

// ---------------------------------------------------------------------------
// Generic 64xC conv as GEMM: Y = W(64,C) * Xin(b,C,P).  C % 4 == 0, P % 64 == 0.
// mode 0: accumulate per-channel sum/sumsq into out[0..63], out[64..127]
// mode 1: store act(a*y+b) to out (b,64,P)
// mode 2: atomic float-max of raw y into out (b,64,P/kred)
// Grid: (P/64, 4, B), 32 threads (1 wave computes 16x64 tile).
__global__ __launch_bounds__(32) void conv_wmma(const float* __restrict__ W,
                                                const float* __restrict__ Xin,
                                                int C, int P, int mode,
                                                const float* __restrict__ asc,
                                                const float* __restrict__ bsh,
                                                int act, float* __restrict__ out,
                                                int kred) {
  const int lane = threadIdx.x & 31;
  const int half = lane >> 4, l16 = lane & 15;
  const int p0 = blockIdx.x * 64;
  const int o0 = blockIdx.y * 16;
  const int bz = blockIdx.z;
  const float* Xb = Xin + (size_t)bz * C * P;
  const v8f vz = {};
  v8f accs[4] = {vz, vz, vz, vz};
  const float* pa = W + (size_t)(o0 + l16) * C + 2 * half;  // 8B-aligned (C%4==0, c even)
  const float* pb = Xb + (size_t)(2 * half) * P + p0 + l16;
  const int ksteps = C >> 2;
  for (int t = 0; t < ksteps; ++t) {
    v2f a = *(const v2f*)pa;
    v2f b0, b1, b2, b3;
    b0.x = pb[0];  b0.y = pb[P];
    b1.x = pb[16]; b1.y = pb[P + 16];
    b2.x = pb[32]; b2.y = pb[P + 32];
    b3.x = pb[48]; b3.y = pb[P + 48];
    accs[0] = __builtin_amdgcn_wmma_f32_16x16x4_f32(false, a, false, b0, (short)0, accs[0], false, false);
    accs[1] = __builtin_amdgcn_wmma_f32_16x16x4_f32(false, a, false, b1, (short)0, accs[1], false, false);
    accs[2] = __builtin_amdgcn_wmma_f32_16x16x4_f32(false, a, false, b2, (short)0, accs[2], false, false);
    accs[3] = __builtin_amdgcn_wmma_f32_16x16x4_f32(false, a, false, b3, (short)0, accs[3], false, false);
    pa += 4;
    pb += (size_t)4 * P;
  }
  if (mode == 0) {
#pragma unroll
    for (int v = 0; v < 8; ++v) {
      float s1 = accs[0][v] + accs[1][v] + accs[2][v] + accs[3][v];
      float s2 = accs[0][v] * accs[0][v] + accs[1][v] * accs[1][v] +
                 accs[2][v] * accs[2][v] + accs[3][v] * accs[3][v];
      for (int off = 1; off < 16; off <<= 1) {
        s1 += __shfl_xor(s1, off, 32);
        s2 += __shfl_xor(s2, off, 32);
      }
      if (l16 == 0) {
        int o = o0 + v + 8 * half;
        atomicAdd(&out[o], s1);
        atomicAdd(&out[64 + o], s2);
      }
    }
  } else if (mode == 1) {
    float* ob = out + (size_t)bz * 64 * P;
#pragma unroll
    for (int nt = 0; nt < 4; ++nt)
#pragma unroll
      for (int v = 0; v < 8; ++v) {
        int o = o0 + v + 8 * half;
        float y = asc[o] * accs[nt][v] + bsh[o];
        y = act ? fmaxf(y, 0.f) : leaky02(y);
        ob[(size_t)o * P + p0 + nt * 16 + l16] = y;
      }
  } else {
    int Nn = P / kred;
    float* ob = out + (size_t)bz * 64 * Nn;
#pragma unroll
    for (int nt = 0; nt < 4; ++nt) {
      int n = (p0 + nt * 16 + l16) / kred;
#pragma unroll
      for (int v = 0; v < 8; ++v) {
        int o = o0 + v + 8 * half;
        atomicMaxF(&ob[(size_t)o * Nn + n], accs[nt][v]);
      }
    }
  }
}

// ---------------------------------------------------------------------------
// BN finalize: a = gamma*rsqrt(var+eps), b = beta - mean*a.  64 threads.
__global__ void bn_finalize(const float* __restrict__ stats,
                            const float* __restrict__ gamma,
                            const float* __restrict__ beta, float count,
                            float* __restrict__ ab) {
  int o = threadIdx.x;
  float mean = stats[o] / count;
  float var = stats[64 + o] / count - mean * mean;
  float a = gamma[o] * rsqrtf(var + 1e-5f);
  ab[o] = a;
  ab[64 + o] = beta[o] - mean * a;
}

// elementwise BN + activation (+optional sigmoid) on (b,64,Nn)
__global__ void bn_act_elem(const float* __restrict__ in, const float* __restrict__ a,
                            const float* __restrict__ bb, int Nn, int act, int dosig,
                            float* __restrict__ out, long total) {
  long t = (long)blockIdx.x * blockDim.x + threadIdx.x;
  if (t >= total) return;
  int o = (int)((t / Nn) & 63);
  float y = a[o] * in[t] + bb[o];
  y = act ? fmaxf(y, 0.f) : leaky02(y);
  if (dosig) y = 1.f / (1.f + __expf(-y));
  out[t] = y;
}

// ---------------------------------------------------------------------------
// top-k (descending, stable) of d[m] = 2*G[n,m] - G[n,n] - G[m,m] per row.
#define KNN_MAX 40
__global__ void knn_topk(const float* __restrict__ G, int Npts, int k,
                         int* __restrict__ idxout) {
  int n = blockIdx.x * blockDim.x + threadIdx.x;
  if (n >= Npts) return;
  float vals[KNN_MAX];
  int ids[KNN_MAX];
  float dnn = G[(size_t)n * Npts + n];
  int cnt = 0;
  for (int m = 0; m < Npts; ++m) {
    float d = 2.f * G[(size_t)n * Npts + m] - dnn - G[(size_t)m * Npts + m];
    if (cnt < k || d > vals[k - 1]) {
      int pos = (cnt < k) ? cnt : k - 1;
      while (pos > 0 && vals[pos - 1] < d) {
        vals[pos] = vals[pos - 1];
        ids[pos] = ids[pos - 1];
        --pos;
      }
      vals[pos] = d;
      ids[pos] = m;
      if (cnt < k) ++cnt;
    }
  }
  for (int i = 0; i < k; ++i) idxout[(size_t)n * k + i] = ids[i];
}

// ---------------------------------------------------------------------------
// graph feature: g(b,Cstore,Npts*k): ch c = X[c,idx]-X[c,n]; ch C+c = X[c,n];
// channels [2C,Cstore) zero-padded (guard-free WMMA consumption).
__global__ void build_gf(const float* __restrict__ X, const int* __restrict__ idx,
                         float* __restrict__ g, int C, int Cstore, int Npts, int k) {
  long t = (long)blockIdx.x * blockDim.x + threadIdx.x;
  int b = blockIdx.y;
  long P = (long)Npts * k;
  if (t >= P) return;
  int n = (int)(t / k);
  int kk = (int)(t % k);
  int j = idx[((size_t)b * Npts + n) * k + kk];
  const float* Xb = X + (size_t)b * C * Npts;
  float* gb = g + (size_t)b * Cstore * P;
  long p = (long)n * k + kk;
  for (int c = 0; c < C; ++c) {
    float ctr = Xb[(size_t)c * Npts + n];
    float ft = Xb[(size_t)c * Npts + j];
    gb[(size_t)c * P + p] = ft - ctr;
    gb[(size_t)(C + c) * P + p] = ctr;
  }
  for (int c = 2 * C; c < Cstore; ++c) gb[(size_t)c * P + p] = 0.f;
}

// ---------------------------------------------------------------------------
// sc1[b,n] = sigmoid( sum_ij h1[b,i,n]*dW[i,j]*X[b,j,n] + db )
__global__ void disc_score(const float* __restrict__ h1, const float* __restrict__ X,
                           const float* __restrict__ dW, const float* __restrict__ db,
                           int Npts, float* __restrict__ s) {
  int n = blockIdx.x * blockDim.x + threadIdx.x;
  int b = blockIdx.y;
  if (n >= Npts) return;
  const float* hb = h1 + (size_t)b * 64 * Npts + n;
  const float* xb = X + (size_t)b * 64 * Npts + n;
  float cv[64];
  for (int j = 0; j < 64; ++j) cv[j] = xb[(size_t)j * Npts];
  float acc = 0.f;
  for (int i = 0; i < 64; ++i) {
    float hi = hb[(size_t)i * Npts];
    const float* wr = dW + i * 64;
    float tj = 0.f;
    for (int j = 0; j < 64; ++j) tj += wr[j] * cv[j];
    acc += hi * tj;
  }
  acc += db[0];
  s[(size_t)b * Npts + n] = 1.f / (1.f + __expf(-acc));
}

// ---------------------------------------------------------------------------
// Per-batch iterative top-m selection (sorted desc, stable). One block / batch.
__global__ void topk_select(const float* __restrict__ s, int Npts, int m,
                            int* __restrict__ selidx, float* __restrict__ selval) {
  __shared__ float sv[2048];
  __shared__ float rmax[256];
  __shared__ int rarg[256];
  int b = blockIdx.x, t = threadIdx.x;
  for (int i = t; i < Npts; i += 256) sv[i] = s[(size_t)b * Npts + i];
  __syncthreads();
  for (int r = 0; r < m; ++r) {
    float best = -3.0e38f;
    int barg = 0x7fffffff;
    for (int i = t; i < Npts; i += 256) {
      float v = sv[i];
      if (v > best || (v == best && i < barg)) { best = v; barg = i; }
    }
    rmax[t] = best;
    rarg[t] = barg;
    __syncthreads();
    for (int off = 128; off > 0; off >>= 1) {
      if (t < off) {
        if (rmax[t + off] > rmax[t] ||
            (rmax[t + off] == rmax[t] && rarg[t + off] < rarg[t])) {
          rmax[t] = rmax[t + off];
          rarg[t] = rarg[t + off];
        }
      }
      __syncthreads();
    }
    if (t == 0) {
      selidx[(size_t)b * m + r] = rarg[0];
      selval[(size_t)b * m + r] = rmax[0];
      sv[rarg[0]] = -3.0e38f;
    }
    __syncthreads();
  }
}

// ---------------------------------------------------------------------------
__global__ void gather_scale(const float* __restrict__ seq, const int* __restrict__ selidx,
                             const float* __restrict__ selval, int Npts, int m,
                             float* __restrict__ fdst, int Cout) {
  int t = blockIdx.x * blockDim.x + threadIdx.x;
  int b = blockIdx.y;
  if (t >= 64 * m) return;
  int o = t / m, r = t % m;
  int id = selidx[(size_t)b * m + r];
  fdst[((size_t)b * Cout + o) * m + r] =
      seq[((size_t)b * 64 + o) * Npts + id] * selval[(size_t)b * m + r];
}

__global__ void gather_nodes(const float* __restrict__ xyz, const int* __restrict__ selidx,
                             int Npts, int m, float* __restrict__ nodes) {
  int t = blockIdx.x * blockDim.x + threadIdx.x;
  int b = blockIdx.y;
  if (t >= 3 * m) return;
  int c = t / m, r = t % m;
  nodes[((size_t)b * 3 + c) * m + r] =
      xyz[((size_t)b * 3 + c) * Npts + selidx[(size_t)b * m + r]];
}

// ---------------------------------------------------------------------------
#define AGG_MAX 20
__global__ void aggregate_k(const float* __restrict__ xyz, int Npts,
                            const float* __restrict__ node, int M,
                            const float* __restrict__ feat, int k,
                            float* __restrict__ fdst, int Cout) {
  int j = blockIdx.x * blockDim.x + threadIdx.x;
  int b = blockIdx.y;
  if (j >= M) return;
  float nx = node[((size_t)b * 3 + 0) * M + j];
  float ny = node[((size_t)b * 3 + 1) * M + j];
  float nz = node[((size_t)b * 3 + 2) * M + j];
  float vals[AGG_MAX];
  int ids[AGG_MAX];
  int cnt = 0;
  const float* xb = xyz + (size_t)b * 3 * Npts;
  for (int n = 0; n < Npts; ++n) {
    float dx = xb[n] - nx, dy = xb[Npts + n] - ny, dz = xb[2 * (size_t)Npts + n] - nz;
    float d = -(dx * dx + dy * dy + dz * dz);
    if (cnt < k || d > vals[k - 1]) {
      int pos = (cnt < k) ? cnt : k - 1;
      while (pos > 0 && vals[pos - 1] < d) {
        vals[pos] = vals[pos - 1];
        ids[pos] = ids[pos - 1];
        --pos;
      }
      vals[pos] = d;
      ids[pos] = n;
      if (cnt < k) ++cnt;
    }
  }
  const float* fb = feat + (size_t)b * 64 * Npts;
  for (int o = 0; o < 64; ++o) {
    float mx = -3.0e38f;
    for (int i = 0; i < k; ++i) mx = fmaxf(mx, fb[(size_t)o * Npts + ids[i]]);
    fdst[((size_t)b * Cout + 64 + o) * M + j] = mx;
  }
}

// ===========================================================================
// Host-side orchestration
// ===========================================================================
static void run_knn(hipStream_t st, const float* X, int C, int Npts, int Bb, int k,
                    float* G, int* idxout) {
  for (int b = 0; b < Bb; ++b) {
    dim3 gg(Npts / 64, Npts / 16);
    gram_wmma<<<gg, 32, 0, st>>>(X + (size_t)b * C * Npts, G, C, Npts);
    knn_topk<<<ceilDiv(Npts, 128), 128, 0, st>>>(G, Npts, k,
                                                 idxout + (size_t)b * Npts * k);
  }
}

static void conv_bn_store(hipStream_t st, const float* W, const float* Xin, int C,
                          int P, int Bb, const float* gamma, const float* beta,
                          int act, float* out, float* stats) {
  fill_f32<<<1, 128, 0, st>>>(stats, 0.f, 128);
  dim3 g(P / 64, 4, Bb);
  conv_wmma<<<g, 32, 0, st>>>(W, Xin, C, P, 0, nullptr, nullptr, 0, stats, 1);
  bn_finalize<<<1, 64, 0, st>>>(stats, gamma, beta, (float)((double)Bb * P),
                                stats + 128);
  conv_wmma<<<g, 32, 0, st>>>(W, Xin, C, P, 1, stats + 128, stats + 192, act, out, 1);
}

static void conv_bn_maxact(hipStream_t st, const float* W, const float* Xin, int C,
                           int P, int Bb, int kred, const float* gamma,
                           const float* beta, int dosig, float* ymax, float* dst,
                           float* stats) {
  fill_f32<<<1, 128, 0, st>>>(stats, 0.f, 128);
  dim3 g(P / 64, 4, Bb);
  conv_wmma<<<g, 32, 0, st>>>(W, Xin, C, P, 0, nullptr, nullptr, 0, stats, 1);
  bn_finalize<<<1, 64, 0, st>>>(stats, gamma, beta, (float)((double)Bb * P),
                                stats + 128);
  int Nn = P / kred;
  long total = (long)Bb * 64 * Nn;
  fill_f32<<<ceilDiv((int)total, 256), 256, 0, st>>>(ymax, -3.0e38f, total);
  conv_wmma<<<g, 32, 0, st>>>(W, Xin, C, P, 2, nullptr, nullptr, 0, ymax, kred);
  bn_act_elem<<<ceilDiv((int)total, 256), 256, 0, st>>>(ymax, stats + 128,
                                                        stats + 192, Nn, 0, dosig,
                                                        dst, total);
}

// full _index_select (seq2/h2/perm are dead code in the reference; fc bias
// cancels under BN over (0,2)).
static void index_select_host(hipStream_t st, const float* xyz, const float* seq,
                              int Npts, int m, int neigh, const float* fcW,
                              const float* fcg, const float* fcbt, const float* cW,
                              const float* cg, const float* cb, const float* dW,
                              const float* db, float* stats, float* G, int* idxbuf,
                              float* h1, float* gbuf, float* ymax, float* Xbuf,
                              float* sbuf, int* selidx, float* selval, float* fdst,
                              int Cout, float* nodesout, int Bb) {
  conv_bn_store(st, fcW, seq, 64, Npts, Bb, fcg, fcbt, /*relu*/ 1, h1, stats);
  run_knn(st, h1, 64, Npts, Bb, neigh, G, idxbuf);
  {
    dim3 g(ceilDiv(Npts * neigh, 256), Bb);
    build_gf<<<g, 256, 0, st>>>(h1, idxbuf, gbuf, 64, 128, Npts, neigh);
  }
  conv_bn_maxact(st, cW, gbuf, 128, Npts * neigh, Bb, neigh, cg, cb, /*sig*/ 1,
                 ymax, Xbuf, stats);
  {
    dim3 g(ceilDiv(Npts, 64), Bb);
    disc_score<<<g, 64, 0, st>>>(h1, Xbuf, dW, db, Npts, sbuf);
  }
  topk_select<<<Bb, 256, 0, st>>>(sbuf, Npts, m, selidx, selval);
  {
    dim3 g(ceilDiv(64 * m, 256), Bb);
    gather_scale<<<g, 256, 0, st>>>(seq, selidx, selval, Npts, m, fdst, Cout);
    dim3 g2(ceilDiv(3 * m, 96), Bb);
    gather_nodes<<<g2, 96, 0, st>>>(xyz, selidx, Npts, m, nodesout);
  }
}

// ===========================================================================
extern "C" void kernel_launch(void* const* d_in, const int* in_sizes, int n_in,
                              void* d_out, int out_size, void* d_ws, size_t ws_size,
                              hipStream_t stream) {
  (void)in_sizes; (void)n_in; (void)out_size; (void)ws_size;
  const int Bb = 16, N1 = 2048, KK = 40;

  const float* x      = (const float*)d_in[0];
  const float* conv1W = (const float*)d_in[2];
  const float* conv2W = (const float*)d_in[3];
  const float* conv3W = (const float*)d_in[4];
  const float* conv4W = (const float*)d_in[5];
  const float* conv5W = (const float*)d_in[6];
  const float* bng    = (const float*)d_in[7];
  const float* bnb    = (const float*)d_in[8];
  const float* fcW    = (const float*)d_in[9];
  const float* fcg    = (const float*)d_in[11];
  const float* fcbt   = (const float*)d_in[12];
  const float* cW     = (const float*)d_in[13];
  const float* cg     = (const float*)d_in[14];
  const float* cb     = (const float*)d_in[15];
  const float* dW     = (const float*)d_in[16];
  const float* db     = (const float*)d_in[17];
  float* out = (float*)d_out;

  // ---- workspace bump allocator --------------------------------------
  size_t off = 0;
  char* base = (char*)d_ws;
  auto allocF = [&](size_t nelem) -> float* {
    float* p = (float*)(base + off);
    off += ((nelem * 4 + 255) & ~(size_t)255);
    return p;
  };
  float* stats  = allocF(256);
  float* Wp     = allocF(512);                        // padded conv1 weights (64x8)
  float* xp     = allocF((size_t)Bb * 4 * N1);        // padded x (b,4,N)
  float* G      = allocF((size_t)N1 * N1);            // 16.8 MB (per-batch reuse)
  int*   idxb   = (int*)allocF((size_t)Bb * N1 * KK); // 5.2 MB
  float* arena1 = allocF((size_t)Bb * 64 * N1 * KK);  // 335 MB (hA/g1/gB/gC/...)
  float* arena2 = allocF((size_t)Bb * 64 * 512 * 20); // 42 MB  (gA(8ch)/hB)
  float* x1     = allocF((size_t)Bb * 64 * N1);
  float* h1     = allocF((size_t)Bb * 64 * N1);
  float* ymax   = allocF((size_t)Bb * 64 * N1);
  float* Xbuf   = allocF((size_t)Bb * 64 * N1);
  float* sbuf   = allocF((size_t)Bb * N1);
  int*   selidx = (int*)allocF((size_t)Bb * 512);
  float* selval = allocF((size_t)Bb * 512);
  float* node1  = allocF((size_t)Bb * 3 * 512);
  float* node2  = allocF((size_t)Bb * 3 * 128);
  float* node3  = allocF((size_t)Bb * 3 * 32);
  float* fB     = allocF((size_t)Bb * 128 * 512);
  float* fC     = allocF((size_t)Bb * 128 * 128);
  float* x2     = allocF((size_t)Bb * 64 * 512);
  float* x3     = allocF((size_t)Bb * 64 * 128);

  // padding prep
  pad_conv1w<<<1, 512, 0, stream>>>(conv1W, Wp);
  {
    dim3 g(ceilDiv(4 * N1, 256), Bb);
    pad_x4<<<g, 256, 0, stream>>>(x, xp, N1);
  }

  // ======================= Stage A (N=2048, k=40) ========================
  run_knn(stream, xp, 4, N1, Bb, KK, G, idxb);
  {
    dim3 g(ceilDiv(N1 * KK, 256), Bb);
    build_gf<<<g, 256, 0, stream>>>(x, idxb, arena2, 3, 8, N1, KK);  // gA (b,8,N*K)
  }
  conv_bn_store(stream, Wp, arena2, 8, N1 * KK, Bb, bng + 0, bnb + 0, 0,
                arena1, stats);
  conv_bn_maxact(stream, conv2W, arena1, 64, N1 * KK, Bb, KK, bng + 64, bnb + 64,
                 0, ymax, x1, stats);
  index_select_host(stream, x, x1, N1, 512, 20, fcW + 0 * 4096, fcg + 0, fcbt + 0,
                    cW + 0 * 8192, cg + 0, cb + 0, dW + 0 * 4096, db + 0, stats, G,
                    idxb, h1, arena1, ymax, Xbuf, sbuf, selidx, selval, fB, 128,
                    node1, Bb);
  {
    dim3 g(ceilDiv(512, 64), Bb);
    aggregate_k<<<g, 64, 0, stream>>>(x, N1, node1, 512, x1, 20, fB, 128);
  }

  // ======================= Stage B (N=512, k=20) =========================
  run_knn(stream, fB, 128, 512, Bb, 20, G, idxb);
  {
    dim3 g(ceilDiv(512 * 20, 256), Bb);
    build_gf<<<g, 256, 0, stream>>>(fB, idxb, arena1, 128, 256, 512, 20);  // gB
  }
  conv_bn_store(stream, conv3W, arena1, 256, 512 * 20, Bb, bng + 128, bnb + 128, 0,
                arena2, stats);  // hB
  conv_bn_maxact(stream, conv4W, arena2, 64, 512 * 20, Bb, 20, bng + 192,
                 bnb + 192, 0, ymax, x2, stats);
  index_select_host(stream, node1, x2, 512, 128, 10, fcW + 1 * 4096, fcg + 64,
                    fcbt + 64, cW + 1 * 8192, cg + 64, cb + 64, dW + 1 * 4096,
                    db + 1, stats, G, idxb, h1, arena1, ymax, Xbuf, sbuf, selidx,
                    selval, fC, 128, node2, Bb);
  {
    dim3 g(ceilDiv(128, 64), Bb);
    aggregate_k<<<g, 64, 0, stream>>>(node1, 512, node2, 128, x2, 10, fC, 128);
  }

  // ======================= Stage C (N=128, k=10) =========================
  run_knn(stream, fC, 128, 128, Bb, 10, G, idxb);
  {
    dim3 g(ceilDiv(128 * 10, 256), Bb);
    build_gf<<<g, 256, 0, stream>>>(fC, idxb, arena1, 128, 256, 128, 10);  // gC
  }
  conv_bn_maxact(stream, conv5W, arena1, 256, 128 * 10, Bb, 10, bng + 256,
                 bnb + 256, 0, ymax, x3, stats);
  index_select_host(stream, node2, x3, 128, 32, 5, fcW + 2 * 4096, fcg + 128,
                    fcbt + 128, cW + 2 * 8192, cg + 128, cb + 128, dW + 2 * 4096,
                    db + 2, stats, G, idxb, h1, arena1, ymax, Xbuf, sbuf, selidx,
                    selval, out, 128, node3, Bb);
  {
    dim3 g(1, Bb);
    aggregate_k<<<g, 64, 0, stream>>>(node2, 128, node3, 32, x3, 5, out, 128);
  }
}